// GraphSageEdgeLayer_77567109366525
// MI455X (gfx1250) — compile-verified
//
#include <hip/hip_runtime.h>
#include <math.h>

typedef float v2f __attribute__((ext_vector_type(2)));
typedef float v8f __attribute__((ext_vector_type(8)));

#define N_NODES 50000
#define N_EDGES 800000
#define DCH 128
#define NT (N_NODES / 16)          // 3125 row tiles of 16 (exact)
#define WPB 4                      // waves per block

// D = A*B + C, A:16x4 f32, B:4x16 f32, C/D:16x16 f32 (wave32)
__device__ __forceinline__ v8f wmma_f32(v2f a, v2f b, v8f c) {
  return __builtin_amdgcn_wmma_f32_16x16x4_f32(
      /*neg_a=*/false, a, /*neg_b=*/false, b,
      /*c_mod=*/(short)0, c, /*reuse_a=*/false, /*reuse_b=*/false);
}

// ---------------------------------------------------------------------------
// Phase 1: Ah = h@A_w + A_b ; Bh = h@B_w + B_b ; zero the segment-max buffer.
// One wave owns a 16-row node tile; A-operand (h) loads are shared between the
// A and B weight streams. Accumulators: 16 tiles x v8f.
// ---------------------------------------------------------------------------
__global__ void __launch_bounds__(128) gemm_ab_kernel(
    const float* __restrict__ h,
    const float* __restrict__ Aw, const float* __restrict__ Ab,
    const float* __restrict__ Bw, const float* __restrict__ Bb,
    float* __restrict__ Ah, float* __restrict__ Bh, float* __restrict__ c) {
  const int lane = threadIdx.x & 31;
  const int wave = threadIdx.x >> 5;
  const int tile = blockIdx.x * WPB + wave;
  if (tile >= NT) return;              // uniform per wave -> EXEC all-ones for WMMA
  const int m0   = tile * 16;
  const int l15  = lane & 15;
  const int koff = (lane >> 4) * 2;    // lanes 16-31 hold K+2,K+3

  // Zero c rows for this tile (atomic-max-uint needs a 0 floor, ws is poisoned).
  float4 z4 = make_float4(0.f, 0.f, 0.f, 0.f);
  float4* c4 = (float4*)(c + (size_t)m0 * DCH);
#pragma unroll
  for (int i = 0; i < 16; ++i) c4[i * 32 + lane] = z4;

  v8f accA[8], accB[8];
#pragma unroll
  for (int t = 0; t < 8; ++t) { accA[t] = v8f{}; accB[t] = v8f{}; }

  const float* hrow = h + (size_t)(m0 + l15) * DCH;

  for (int k0 = 0; k0 < DCH; k0 += 4) {
    v2f a;
    a.x = hrow[k0 + koff];
    a.y = hrow[k0 + koff + 1];
    const float* awr = Aw + (size_t)(k0 + koff) * DCH;
    const float* bwr = Bw + (size_t)(k0 + koff) * DCH;
#pragma unroll
    for (int t = 0; t < 8; ++t) {
      const int col = t * 16 + l15;
      v2f ba, bb;
      ba.x = awr[col];       ba.y = awr[DCH + col];
      bb.x = bwr[col];       bb.y = bwr[DCH + col];
      accA[t] = wmma_f32(a, ba, accA[t]);
      accB[t] = wmma_f32(a, bb, accB[t]);
    }
  }

  const int rowAdd = (lane >> 4) * 8;  // VGPR r: lanes0-15 -> M=r, lanes16-31 -> M=r+8
#pragma unroll
  for (int t = 0; t < 8; ++t) {
    const int col = t * 16 + l15;
    const float ab = Ab[col], bbias = Bb[col];
#pragma unroll
    for (int r = 0; r < 8; ++r) {
      const size_t row = (size_t)(m0 + rowAdd + r);
      Ah[row * DCH + col] = accA[t][r] + ab;
      Bh[row * DCH + col] = accB[t][r] + bbias;
    }
  }
}

// ---------------------------------------------------------------------------
// Phase 2: per-edge gate + relu + segment-max (uint atomicMax on non-negative
// float bit patterns == exact float max, order independent / deterministic).
// One wave per edge; lane handles a float4 slice (32*4 = 128 channels).
// ---------------------------------------------------------------------------
__global__ void __launch_bounds__(256) edge_kernel(
    const float* __restrict__ Ah, const float* __restrict__ Bh,
    const int* __restrict__ src, const int* __restrict__ dst,
    unsigned int* __restrict__ cbits) {
  const int lane = threadIdx.x & 31;
  const int wave = threadIdx.x >> 5;
  const int e = blockIdx.x * 8 + wave;
  if (e >= N_EDGES) return;
  const int s = src[e];
  const int d = dst[e];

  const float4 bs = *(const float4*)(Bh + (size_t)s * DCH + lane * 4);
  const float4 bd = *(const float4*)(Bh + (size_t)d * DCH + lane * 4);
  const float4 as = *(const float4*)(Ah + (size_t)s * DCH + lane * 4);
  unsigned int* cw = cbits + (size_t)d * DCH + lane * 4;

  float gsum[4] = {bs.x + bd.x, bs.y + bd.y, bs.z + bd.z, bs.w + bd.w};
  float av[4]   = {as.x, as.y, as.z, as.w};
#pragma unroll
  for (int j = 0; j < 4; ++j) {
    float g = av[j] / (1.0f + __expf(-gsum[j]));  // sigmoid gate
    g = fmaxf(g, 0.0f);                           // relu -> non-negative
    atomicMax(cw + j, __float_as_uint(g));
  }
}

// ---------------------------------------------------------------------------
// Phase 3: bundle = [h, c] @ U_w + U_b (K=256 WMMA), row L2-normalize, relu,
// residual. Bundle tile staged in LDS (rows padded to 132 floats).
// ---------------------------------------------------------------------------
__global__ void __launch_bounds__(128) update_kernel(
    const float* __restrict__ h, const float* __restrict__ c,
    const float* __restrict__ Uw, const float* __restrict__ Ub,
    float* __restrict__ out) {
  __shared__ __align__(16) float lds[WPB][16][132];
  __shared__ float rinv[WPB][16];

  const int lane = threadIdx.x & 31;
  const int wave = threadIdx.x >> 5;
  const int tile = blockIdx.x * WPB + wave;
  if (tile >= NT) return;
  const int m0   = tile * 16;
  const int l15  = lane & 15;
  const int koff = (lane >> 4) * 2;

  v8f acc[8];
#pragma unroll
  for (int t = 0; t < 8; ++t) acc[t] = v8f{};

  const float* hrow = h + (size_t)(m0 + l15) * DCH;
  const float* crow = c + (size_t)(m0 + l15) * DCH;

  for (int k0 = 0; k0 < 2 * DCH; k0 += 4) {
    const float* X = (k0 < DCH) ? hrow : crow;   // concat(h, c) along K
    const int kk   = (k0 < DCH) ? k0 : (k0 - DCH);
    v2f a;
    a.x = X[kk + koff];
    a.y = X[kk + koff + 1];
    const float* uw = Uw + (size_t)(k0 + koff) * DCH;
#pragma unroll
    for (int t = 0; t < 8; ++t) {
      const int col = t * 16 + l15;
      v2f b;
      b.x = uw[col];
      b.y = uw[DCH + col];
      acc[t] = wmma_f32(a, b, acc[t]);
    }
  }

  const int rowAdd = (lane >> 4) * 8;
#pragma unroll
  for (int t = 0; t < 8; ++t) {
    const int col = t * 16 + l15;
    const float ub = Ub[col];
#pragma unroll
    for (int r = 0; r < 8; ++r)
      lds[wave][rowAdd + r][col] = acc[t][r] + ub;
  }
  __builtin_amdgcn_wave_barrier();

  // Row L2 norm: 2 lanes per row, each sums 64 squares, combine across halves.
  const int half = lane >> 4;
  float s = 0.0f;
#pragma unroll 8
  for (int i = 0; i < 64; ++i) {
    float v = lds[wave][l15][half * 64 + i];
    s += v * v;
  }
  s += __shfl_xor(s, 16);
  if (half == 0) rinv[wave][l15] = 1.0f / fmaxf(sqrtf(s), 1e-12f);
  __builtin_amdgcn_wave_barrier();

  // out = h + relu(bundle / norm); one full row (128 ch) per iteration.
#pragma unroll 4
  for (int j = 0; j < 16; ++j) {
    const float inv = rinv[wave][j];
    const float4 b4 = *(const float4*)&lds[wave][j][lane * 4];
    const size_t gi = ((size_t)(m0 + j) * DCH + lane * 4);
    const float4 h4 = *(const float4*)(h + gi);
    float4 o4;
    o4.x = h4.x + fmaxf(b4.x * inv, 0.0f);
    o4.y = h4.y + fmaxf(b4.y * inv, 0.0f);
    o4.z = h4.z + fmaxf(b4.z * inv, 0.0f);
    o4.w = h4.w + fmaxf(b4.w * inv, 0.0f);
    *(float4*)(out + gi) = o4;
  }
}

// ---------------------------------------------------------------------------
extern "C" void kernel_launch(void* const* d_in, const int* in_sizes, int n_in,
                              void* d_out, int out_size, void* d_ws, size_t ws_size,
                              hipStream_t stream) {
  const float* h   = (const float*)d_in[0];
  const int*   src = (const int*)d_in[1];
  const int*   dst = (const int*)d_in[2];
  const float* Aw  = (const float*)d_in[3];
  const float* Ab  = (const float*)d_in[4];
  const float* Bw  = (const float*)d_in[5];
  const float* Bb  = (const float*)d_in[6];
  const float* Uw  = (const float*)d_in[7];
  const float* Ub  = (const float*)d_in[8];
  float* out = (float*)d_out;

  // Workspace: Ah | Bh | c  (3 x 25.6 MB, all L2-resident on MI455X)
  float* Ah = (float*)d_ws;
  float* Bh = Ah + (size_t)N_NODES * DCH;
  float* c  = Bh + (size_t)N_NODES * DCH;

  const dim3 blk(128);
  const dim3 grd((NT + WPB - 1) / WPB);

  gemm_ab_kernel<<<grd, blk, 0, stream>>>(h, Aw, Ab, Bw, Bb, Ah, Bh, c);
  edge_kernel<<<dim3((N_EDGES + 7) / 8), dim3(256), 0, stream>>>(
      Ah, Bh, src, dst, (unsigned int*)c);
  update_kernel<<<grd, blk, 0, stream>>>(h, c, Uw, Ub, out);
}